// DetectionLayer_21174188769536
// MI455X (gfx1250) — compile-verified
//
#include <hip/hip_runtime.h>
#include <cstdint>
#include <cstddef>

// ---------------- problem constants (from reference) ----------------
#define BB   16      // batch
#define NN   1000    // rois per batch
#define CC   81      // classes
#define KK   100     // max detections
#define FF   1024    // feature size
#define NPAD 1024    // padded N for bitonic sort
#define META_STRIDE (12 + CC)   // 93
#define MIN_CONF 0.7f
#define NMS_THR  0.3f

#define INF_F (__builtin_inff())

// ---- gfx1250 async global<->LDS path (guarded; fallback = float4 copy) ----
#if defined(__HIP_DEVICE_COMPILE__)
#if __has_builtin(__builtin_amdgcn_global_load_async_to_lds_b128) && \
    __has_builtin(__builtin_amdgcn_global_store_async_from_lds_b128) && \
    __has_builtin(__builtin_amdgcn_s_wait_asynccnt)
#define USE_ASYNC_LDS 1
#endif
#endif

#ifdef USE_ASYNC_LDS
// Exact types from clang's diagnostic: pointer to vector_size(16) int,
// AS1 (global) for the memory side, AS3 (shared) for the LDS side.
typedef int v4i_t __attribute__((vector_size(16)));
typedef v4i_t __attribute__((address_space(1)))* gbl_v4i_p;
typedef v4i_t __attribute__((address_space(3)))* lds_v4i_p;
#endif

// =====================================================================
// Feature gather (first in file so the disasm snippet shows it):
// one block per output row (B*K blocks, 256 threads). Branch-free:
// the NMS kernel pre-resolved each row's source pointer (kept feature
// row, or a zeroed scratch row). Pure async global->LDS->global DMA
// tracked by ASYNCcnt.
// =====================================================================
__global__ __launch_bounds__(256) void det_gather_kernel(
    const unsigned long long* __restrict__ ws_src,  // [B*K] source ptrs
    float*                    __restrict__ feat_out) // [B,K,F]
{
    const int row = (int)blockIdx.x;      // 0 .. B*K-1
    const int tid = (int)threadIdx.x;
    const float* s   = (const float*)(uintptr_t)ws_src[row]; // uniform -> s_load
    float*       dst = feat_out + (size_t)row * FF;

#ifdef USE_ASYNC_LDS
    __shared__ float sbuf[FF];
    // each lane DMAs 16B global -> LDS without touching the VGPR data path
    __builtin_amdgcn_global_load_async_to_lds_b128(
        (gbl_v4i_p)(s + tid * 4),
        (lds_v4i_p)(&sbuf[tid * 4]), 0, 0);
    // wave-local dependency: this wave's loads have fully landed in LDS
    __builtin_amdgcn_s_wait_asynccnt(0);
    // LDS -> global, also via the async engine; each wave stores exactly
    // the LDS region its own lanes loaded, so no block barrier is needed.
    __builtin_amdgcn_global_store_async_from_lds_b128(
        (gbl_v4i_p)(dst + tid * 4),
        (lds_v4i_p)(&sbuf[tid * 4]), 0, 0);
    __builtin_amdgcn_s_wait_asynccnt(0);
#else
    ((float4*)dst)[tid] = ((const float4*)s)[tid];
#endif
}

// =====================================================================
// Kernel 1: per-ROI argmax over classes, delta decode, clip.
// One wave32 per ROI.  grid = B*N waves.
// =====================================================================
__global__ __launch_bounds__(256) void det_refine_kernel(
    const float* __restrict__ rois,       // [B,N,4]
    const float* __restrict__ probs,      // [B,N,C]
    const float* __restrict__ deltas,     // [B,N,C,4]
    const float* __restrict__ meta,       // [B, 12+C]
    int*         __restrict__ ws_cls,     // [B*N]
    float*       __restrict__ ws_score,   // [B*N]
    float*       __restrict__ ws_box)     // [B*N,4]
{
    const int gw   = (int)((blockIdx.x * blockDim.x + threadIdx.x) >> 5);
    const int lane = (int)(threadIdx.x & 31);
    if (gw >= BB * NN) return;
    const int b = gw / NN;
    const int n = gw % NN;
    const size_t base = (size_t)b * NN + n;

    // lane-strided argmax over C=81 (first-max-index semantics)
    const float* p = probs + base * CC;
    float bv = -1.0f;
    int   bc = 0;
    for (int c = lane; c < CC; c += 32) {
        float v = p[c];
        if (v > bv) { bv = v; bc = c; }
    }
    for (int off = 16; off > 0; off >>= 1) {
        float ov = __shfl_down(bv, off, 32);
        int   oc = __shfl_down(bc, off, 32);
        if (ov > bv || (ov == bv && oc < bc)) { bv = ov; bc = oc; }
    }

    if (lane == 0) {
        // gather winning deltas, apply BBOX_STD
        const float* d = deltas + (base * CC + bc) * 4;
        float d0 = d[0] * 0.1f, d1 = d[1] * 0.1f;
        float d2 = d[2] * 0.2f, d3 = d[3] * 0.2f;
        const float* r = rois + base * 4;
        float y1 = r[0], x1 = r[1], y2 = r[2], x2 = r[3];
        float hh = y2 - y1, ww = x2 - x1;
        float cy = y1 + 0.5f * hh + d0 * hh;
        float cx = x1 + 0.5f * ww + d1 * ww;
        hh *= expf(d2);
        ww *= expf(d3);
        float ny1 = cy - 0.5f * hh, nx1 = cx - 0.5f * ww;
        float ny2 = ny1 + hh,       nx2 = nx1 + ww;
        // window (scale from image_meta[0], window row per batch)
        float H  = meta[4], W = meta[5];
        float sy = H - 1.0f, sx = W - 1.0f;
        const float* mb = meta + (size_t)b * META_STRIDE;
        float wy1 =  mb[7]         / sy;
        float wx1 =  mb[8]         / sx;
        float wy2 = (mb[9]  - 1.f) / sy;
        float wx2 = (mb[10] - 1.f) / sx;
        ny1 = fminf(fmaxf(ny1, wy1), wy2);
        nx1 = fminf(fmaxf(nx1, wx1), wx2);
        ny2 = fminf(fmaxf(ny2, wy1), wy2);
        nx2 = fminf(fmaxf(nx2, wx1), wx2);

        ws_cls[base]   = bc;
        ws_score[base] = bv;
        float4* bo = (float4*)ws_box;
        bo[base] = make_float4(ny1, nx1, ny2, nx2);
    }
}

// =====================================================================
// Kernel 2: per-batch bitonic argsort + greedy class-aware NMS (wave 0)
// + det output + source-pointer table + L2 prefetch of kept rows.
// grid = B blocks of 1024 threads.
// =====================================================================
__global__ __launch_bounds__(1024) void det_nms_kernel(
    const int*   __restrict__ ws_cls,
    const float* __restrict__ ws_score,
    const float* __restrict__ ws_box,
    const float* __restrict__ obj_feat,   // [B,N,F]
    unsigned long long* __restrict__ ws_src, // [B*K] resolved src ptrs
    float*       __restrict__ zrow,       // [F] scratch zero row
    float*       __restrict__ det_out)    // [B,K,6]
{
    __shared__ float  s_key[NPAD];
    __shared__ int    s_idx[NPAD];
    __shared__ float4 s_box[NPAD];
    __shared__ int    s_cls[NPAD];
    __shared__ float  s_score[NPAD];
    __shared__ int    s_keep[NPAD];   // kept sorted-positions, in order
    __shared__ int    s_counts[CC];
    __shared__ int    s_nkept;

    const int b   = (int)blockIdx.x;
    const int tid = (int)threadIdx.x;
    const size_t base = (size_t)b * NN;

    // refresh the shared zero row every call (ws is not re-poisoned;
    // all blocks write the same zeros -> benign overlap)
    for (int t = tid; t < FF; t += 1024) zrow[t] = 0.0f;

    // ---- load per-ROI data, build sort keys ----
    if (tid < NN) {
        int   ci = ws_cls[base + tid];
        float sc = ws_score[base + tid];
        s_cls[tid]   = ci;
        s_score[tid] = sc;
        s_box[tid]   = ((const float4*)ws_box)[base + tid];
        bool valid   = (ci > 0) && (sc >= MIN_CONF);
        s_key[tid]   = valid ? -sc : INF_F;
    } else {
        s_cls[tid]   = 0;
        s_score[tid] = 0.0f;
        s_box[tid]   = make_float4(0.f, 0.f, 0.f, 0.f);
        s_key[tid]   = INF_F;
    }
    s_idx[tid] = tid;
    if (tid < CC) s_counts[tid] = 0;
    if (tid == 0) s_nkept = 0;
    __syncthreads();

    // ---- bitonic sort ascending on key (valid boxes first, by -score) ----
    for (unsigned k = 2; k <= NPAD; k <<= 1) {
        for (unsigned j = k >> 1; j > 0; j >>= 1) {
            unsigned i   = (unsigned)tid;
            unsigned ixj = i ^ j;
            if (ixj > i) {
                bool  up = ((i & k) == 0);
                float ki = s_key[i], kj = s_key[ixj];
                if ((ki > kj) == up) {
                    s_key[i] = kj; s_key[ixj] = ki;
                    int t = s_idx[i]; s_idx[i] = s_idx[ixj]; s_idx[ixj] = t;
                }
            }
            __syncthreads();
        }
    }

    // ---- greedy NMS on wave 0 (lanes parallelize over kept list) ----
    if (tid < 32) {
        const int lane = tid;
        int nKept = 0;
        for (int p = 0; p < NPAD; ++p) {
            float kp = s_key[p];
            if (kp == INF_F) break;            // invalid/padding start here
            int    oi = s_idx[p];
            float4 bi = s_box[oi];
            int    ci = s_cls[oi];
            float  ai = (bi.z - bi.x) * (bi.w - bi.y);
            bool sup = false;
            for (int r = lane; r < nKept; r += 32) {
                int oj = s_idx[s_keep[r]];
                if (s_cls[oj] == ci) {
                    float4 bj = s_box[oj];
                    float aj  = (bj.z - bj.x) * (bj.w - bj.y);
                    float yy1 = fmaxf(bi.x, bj.x);
                    float xx1 = fmaxf(bi.y, bj.y);
                    float yy2 = fminf(bi.z, bj.z);
                    float xx2 = fminf(bi.w, bj.w);
                    float inter = fmaxf(yy2 - yy1, 0.f) * fmaxf(xx2 - xx1, 0.f);
                    float uni   = fmaxf(ai + aj - inter, 1e-8f);
                    if (inter > NMS_THR * uni) sup = true;
                }
            }
            int any = __any(sup);
            int app = 0;
            if (lane == 0 && !any && s_counts[ci] < KK) {
                s_counts[ci]++;
                s_keep[nKept] = p;
                app = 1;
            }
            app = __shfl(app, 0, 32);
            nKept += app;
        }
        if (lane == 0) s_nkept = nKept;
    }
    __syncthreads();
    const int nK = s_nkept;

    // ---- det output: [y1,x1,y2,x2,class,score], zero past nK ----
    for (int t = tid; t < KK * 6; t += 1024) {
        int r = t / 6, c6 = t % 6;
        float v = 0.0f;
        if (r < nK) {
            int oi = s_idx[s_keep[r]];
            float4 bx = s_box[oi];
            if      (c6 == 0) v = bx.x;
            else if (c6 == 1) v = bx.y;
            else if (c6 == 2) v = bx.z;
            else if (c6 == 3) v = bx.w;
            else if (c6 == 4) v = (float)s_cls[oi];
            else              v = s_score[oi];
        }
        det_out[((size_t)b * KK) * 6 + t] = v;
    }

    // ---- resolve source pointer per output row (kept row or zero row) ----
    for (int r = tid; r < KK; r += 1024) {
        const float* p = (r < nK)
            ? (obj_feat + ((size_t)b * NN + s_idx[s_keep[r]]) * FF)
            : zrow;
        ws_src[b * KK + r] = (unsigned long long)(uintptr_t)p;
    }

    // ---- prefetch kept feature rows into L2 (global_prefetch_b8) ----
    for (int t = tid; t < KK * 16; t += 1024) {
        int r = t >> 4, line = t & 15;
        if (r < nK) {
            int src = s_idx[s_keep[r]];
            __builtin_prefetch(obj_feat + ((size_t)b * NN + src) * FF + line * 64, 0, 1);
        }
    }
}

// =====================================================================
// Host launcher
// =====================================================================
extern "C" void kernel_launch(void* const* d_in, const int* in_sizes, int n_in,
                              void* d_out, int out_size, void* d_ws, size_t ws_size,
                              hipStream_t stream) {
    const float* rois     = (const float*)d_in[0];   // [B,N,4]
    const float* fpn_cls  = (const float*)d_in[1];   // [B,N,C]
    const float* fpn_bbox = (const float*)d_in[2];   // [B,N,C,4]
    const float* obj_feat = (const float*)d_in[3];   // [B,N,1,1,F]
    const float* meta     = (const float*)d_in[4];   // [B,12+C]

    float* det_out  = (float*)d_out;                   // B*K*6
    float* feat_out = det_out + (size_t)BB * KK * 6;   // B*K*F

    // workspace layout (all 16B aligned)
    char*  ws       = (char*)d_ws;
    int*   ws_cls   = (int*)(ws + 0);        //  64000 B
    float* ws_score = (float*)(ws + 64000);  //  64000 B
    float* ws_box   = (float*)(ws + 128000); // 256000 B
    unsigned long long* ws_src = (unsigned long long*)(ws + 384000); // 12800 B
    float* zrow     = (float*)(ws + 396800); //   4096 B zero row

    // 1) per-ROI refine: one wave32 per ROI, 8 waves per 256-thread block
    {
        int waves  = BB * NN;                // 16000
        int blocks = (waves * 32 + 255) / 256;
        det_refine_kernel<<<blocks, 256, 0, stream>>>(
            rois, fpn_cls, fpn_bbox, meta, ws_cls, ws_score, ws_box);
    }
    // 2) sort + NMS + det output + source pointer table per batch
    det_nms_kernel<<<BB, 1024, 0, stream>>>(
        ws_cls, ws_score, ws_box, obj_feat, ws_src, zrow, det_out);
    // 3) feature gather (branch-free async LDS DMA path)
    det_gather_kernel<<<BB * KK, 256, 0, stream>>>(ws_src, feat_out);
}